// EventWarping_5961414607473
// MI455X (gfx1250) — compile-verified
//
#include <hip/hip_runtime.h>
#include <stdint.h>

// Problem constants (fixed by the reference).
#define BQ   8
#define NQ   200000
#define HQ   480
#define WQ   640
#define HWQ  (HQ * WQ)
#define FSC  640.0f

typedef __attribute__((ext_vector_type(2))) float v2f;
typedef __attribute__((ext_vector_type(8))) float v8f;

// ---------------------------------------------------------------------------
// Helpers
// ---------------------------------------------------------------------------

__device__ __forceinline__ void afadd(float* p, float v) {
  // relaxed, agent scope -> lowers to global_atomic_add_f32 on gfx1250
  __hip_atomic_fetch_add(p, v, __ATOMIC_RELAXED, __HIP_MEMORY_SCOPE_AGENT);
}

// Wave32 sum via the matrix pipe: D = A x ones(4x16).
// A layout (f32 16x4): lane m (0-15): vgpr0->(m,0), vgpr1->(m,1);
// lane m+16: vgpr0->(m,2), vgpr1->(m,3). With a = {v, 0}:
// rowsum_m = v_m + v_{m+16}; D[m,n] = rowsum_m. D layout: lane n (0-15)
// vgpr r = rowsum_r, lane n+16 vgpr r = rowsum_{r+8}. Sum own 8 vgprs,
// then one shfl_xor(16) -> every lane holds the full 32-lane sum.
// Requires EXEC all-ones (call from uniform control flow only).
__device__ __forceinline__ float waveReduceWmma(float v) {
  v2f a; a.x = v; a.y = 0.0f;
  v2f b; b.x = 1.0f; b.y = 1.0f;
  v8f c = {};
  c = __builtin_amdgcn_wmma_f32_16x16x4_f32(
      /*neg_a=*/false, a, /*neg_b=*/false, b,
      /*c_mod=*/(short)0, c, /*reuse_a=*/false, /*reuse_b=*/false);
  float s = ((c[0] + c[1]) + (c[2] + c[3])) + ((c[4] + c[5]) + (c[6] + c[7]));
  s += __shfl_xor(s, 16, 32);
  return s;  // valid in all lanes
}

__device__ __forceinline__ float blockReduceSum(float v) {
  __shared__ float red[8];  // 256 threads / wave32 = 8 waves
  const float ws = waveReduceWmma(v);  // uniform point: EXEC all ones
  const int lane = threadIdx.x & 31;
  const int wid  = threadIdx.x >> 5;
  if (lane == 0) red[wid] = ws;
  __syncthreads();
  float s = 0.0f;
  if (wid == 0) {
    s = (lane < 8) ? red[lane] : 0.0f;
    s += __shfl_down(s, 4, 32);
    s += __shfl_down(s, 2, 32);
    s += __shfl_down(s, 1, 32);
  }
  return s;  // valid in thread 0
}

// ---------------------------------------------------------------------------
// Kernel 1: zero the accumulators (B*2 phases * HW pixels * 4 planes) + d_out
// ---------------------------------------------------------------------------
__global__ __launch_bounds__(256) void zero_ws_kernel(float4* __restrict__ ws,
                                                      float* __restrict__ out,
                                                      int n4) {
  for (int i = blockIdx.x * blockDim.x + threadIdx.x; i < n4;
       i += gridDim.x * blockDim.x) {
    ws[i] = make_float4(0.f, 0.f, 0.f, 0.f);
  }
  if (blockIdx.x == 0 && threadIdx.x == 0) out[0] = 0.0f;
}

// ---------------------------------------------------------------------------
// Kernel 2: fused two-phase bilinear scatter with async global->LDS staging.
// Accumulator layout: acc[((b*2+phase)*HW + pix)*4 + c],
//   c: 0 = pos count, 1 = pos*ts, 2 = neg count, 3 = neg*ts
// so the two atomics per corner hit adjacent addresses (same L2 sector).
// ---------------------------------------------------------------------------
__device__ __forceinline__ void scatterCorner(float* __restrict__ ap, int y,
                                              int x, float w, float tsw,
                                              int cb) {
  if (y >= 0 && y < HQ && x >= 0 && x < WQ && w != 0.0f) {
    float* p = ap + (((size_t)(y * WQ + x)) << 2) + cb;
    afadd(p, w);
    afadd(p + 1, w * tsw);
  }
}

__global__ __launch_bounds__(256) void scatter_kernel(
    const float* __restrict__ flow, const float4* __restrict__ events,
    float* __restrict__ acc) {
  __shared__ float4 ebuf[2][256];  // double buffer, 8 KB

  const int b   = blockIdx.y;
  const int tid = threadIdx.x;
  const float* flowb = flow + (size_t)b * 2 * HWQ;
  float* accb = acc + (size_t)b * 2 * (size_t)HWQ * 4;
  const unsigned long long evbase =
      (unsigned long long)(events + (size_t)b * NQ);

  const int numChunks = (NQ + 255) >> 8;  // 782
  const int stride    = gridDim.x;
  const int c0        = blockIdx.x;
  if (c0 >= numChunks) return;

  // Per-lane async copy of this lane's event (16B) into LDS. One instruction
  // per wave per chunk; tracked with ASYNCcnt (per-wave, in-order).
  auto issue = [&](int c, int buf) {
    int idx = c * 256 + tid;
    if (idx > NQ - 1) idx = NQ - 1;  // clamp tail (harmless duplicate copy)
    unsigned goff = (unsigned)idx * 16u;
    unsigned loff = (unsigned)(uintptr_t)(&ebuf[buf][tid]);
    asm volatile("global_load_async_to_lds_b128 %0, %1, %2"
                 :: "v"(loff), "v"(goff), "s"(evbase)
                 : "memory");
  };

  issue(c0, 0);
  int buf = 0;
  for (int c = c0; c < numChunks; c += stride) {
    const int next = c + stride;
    if (next < numChunks) {
      issue(next, buf ^ 1);
      asm volatile("s_wait_asynccnt 0x1" ::: "memory");  // current chunk ready
    } else {
      asm volatile("s_wait_asynccnt 0x0" ::: "memory");
    }

    const float4 e = ebuf[buf][tid];  // (ts, y, x, p*2-1)
    const int idx = c * 256 + tid;
    if (idx < NQ) {
      const float ts = e.x;
      const int iy = (int)e.y;
      const int ix = (int)e.z;
      const int gp = iy * WQ + ix;
      const float fx = flowb[gp];        // channel 0: x-flow
      const float fy = flowb[HWQ + gp];  // channel 1: y-flow
      const int cb = (e.w > 0.0f) ? 0 : 2;  // polarity plane select

      #pragma unroll
      for (int ph = 0; ph < 2; ++ph) {
        const float tref = (ph == 0) ? 1.0f : 0.0f;
        const float tsw  = (ph == 0) ? ts : (1.0f - ts);
        const float dt   = tref - ts;
        const float wy = e.y + dt * fy * FSC;
        const float wx = e.z + dt * fx * FSC;
        const float ty = floorf(wy), lx = floorf(wx);
        const int ity = (int)ty, ilx = (int)lx;
        const float fyv = wy - ty, fxv = wx - lx;
        float* ap = accb + (size_t)ph * HWQ * 4;
        scatterCorner(ap, ity,     ilx,     (1.0f - fyv) * (1.0f - fxv), tsw, cb);
        scatterCorner(ap, ity,     ilx + 1, (1.0f - fyv) * fxv,          tsw, cb);
        scatterCorner(ap, ity + 1, ilx,     fyv * (1.0f - fxv),          tsw, cb);
        scatterCorner(ap, ity + 1, ilx + 1, fyv * fxv,                   tsw, cb);
      }
    }
    buf ^= 1;
  }
}

// ---------------------------------------------------------------------------
// Kernel 3: per-pixel normalize + square, global sum.
// term = (posn/(pos+1e-9))^2 + (negn/(neg+1e-9))^2 over both phases.
// ---------------------------------------------------------------------------
__global__ __launch_bounds__(256) void reduce_iwe_kernel(
    const float4* __restrict__ acc, float* __restrict__ out, int n4) {
  float s = 0.0f;
  for (int i = blockIdx.x * blockDim.x + threadIdx.x; i < n4;
       i += gridDim.x * blockDim.x) {
    const float4 v = acc[i];
    const float a = v.y / (v.x + 1e-9f);
    const float b = v.w / (v.z + 1e-9f);
    s += a * a + b * b;
  }
  s = blockReduceSum(s);
  if (threadIdx.x == 0) afadd(out, s);
}

// ---------------------------------------------------------------------------
// Kernel 4: Charbonnier smoothness on flow, global sum.
// ---------------------------------------------------------------------------
__global__ __launch_bounds__(256) void smooth_kernel(
    const float* __restrict__ flow, float* __restrict__ out) {
  const int n = BQ * 2 * HWQ;
  float s = 0.0f;
  for (int i = blockIdx.x * blockDim.x + threadIdx.x; i < n;
       i += gridDim.x * blockDim.x) {
    const int w  = i % WQ;
    const int hw = i % HWQ;
    const int h  = hw / WQ;
    const float f0 = flow[i];
    if (h < HQ - 1) {
      const float d = f0 - flow[i + WQ];
      s += sqrtf(d * d + 1e-6f);
    }
    if (w < WQ - 1) {
      const float d = f0 - flow[i + 1];
      s += sqrtf(d * d + 1e-6f);
    }
  }
  s = blockReduceSum(s);
  if (threadIdx.x == 0) afadd(out, s);
}

// ---------------------------------------------------------------------------
// Launch
// ---------------------------------------------------------------------------
extern "C" void kernel_launch(void* const* d_in, const int* in_sizes, int n_in,
                              void* d_out, int out_size, void* d_ws,
                              size_t ws_size, hipStream_t stream) {
  const float*  flow   = (const float*)d_in[0];   // [B,2,H,W]
  const float4* events = (const float4*)d_in[1];  // [B,N,4]
  // d_in[2] (pol_mask) is derivable from event_list[...,3]; unused.
  float* out = (float*)d_out;
  float* acc = (float*)d_ws;  // needs B*2*HW*4 floats = 78.6 MB

  const int n4 = BQ * 2 * HWQ;  // number of float4 accumulator cells

  zero_ws_kernel<<<4096, 256, 0, stream>>>((float4*)d_ws, out, n4);
  scatter_kernel<<<dim3(128, BQ), 256, 0, stream>>>(flow, events, acc);
  reduce_iwe_kernel<<<4096, 256, 0, stream>>>((const float4*)d_ws, out, n4);
  smooth_kernel<<<2048, 256, 0, stream>>>(flow, out);
}